// MultiBoxLoss_29257317220888
// MI455X (gfx1250) — compile-verified
//
#include <hip/hip_runtime.h>

#define NPRI 8732
#define NB   32
#define NOBJ 16
#define NCLS 81
#define NROWS (NB * NPRI)                 // 279424
#define NGROUPS (NROWS / 16)              // 17464 (exact)
#define CLS_BLOCKS (NGROUPS / 8)          // 2183 (exact, 8 waves/block)
#define SORT_N 16384

typedef float v2f __attribute__((ext_vector_type(2)));
typedef float v8f __attribute__((ext_vector_type(8)));

// ---------------------------------------------------------------- init
__global__ void k_init(float* fsc, unsigned* usc) {
    if (threadIdx.x < 3) fsc[threadIdx.x] = 0.0f;   // abs_sum, cls_pos, cls_hard
    if (threadIdx.x == 3) usc[0] = 0u;              // n_pos
}

// ---------------------------------------------------------------- matching
__global__ void k_match(const float* __restrict__ pred_boxes,
                        const float* __restrict__ true_boxes,
                        const int*   __restrict__ true_classes,
                        const float* __restrict__ pboxes,
                        float* __restrict__ ov_g,
                        int*   __restrict__ idx_g,
                        int*   __restrict__ labels_g,
                        float* __restrict__ fsc,
                        unsigned* __restrict__ usc)
{
    __shared__ float s_cc[NOBJ][4];
    __shared__ float s_bc[NOBJ][4];
    __shared__ int   s_tc[NOBJ];
    __shared__ unsigned long long s_best[NOBJ];
    __shared__ float s_rf[256];
    __shared__ int   s_ri[256];

    const int b   = blockIdx.x;
    const int tid = threadIdx.x;

    if (tid < NOBJ) {
        const float* tb = true_boxes + (b * NOBJ + tid) * 4;
        float x = tb[0], y = tb[1], w = tb[2], h = tb[3];
        float cx = (x + w * 0.5f) / 300.0f;
        float cy = (y + h * 0.5f) / 300.0f;
        float cw = w / 300.0f;
        float ch = h / 300.0f;
        s_cc[tid][0] = cx; s_cc[tid][1] = cy; s_cc[tid][2] = cw; s_cc[tid][3] = ch;
        s_bc[tid][0] = cx - cw * 0.5f; s_bc[tid][1] = cy - ch * 0.5f;
        s_bc[tid][2] = cx + cw * 0.5f; s_bc[tid][3] = cy + ch * 0.5f;
        s_tc[tid] = true_classes[b * NOBJ + tid];
        s_best[tid] = 0ull;
    }
    __syncthreads();

    unsigned long long best[NOBJ];
    #pragma unroll
    for (int o = 0; o < NOBJ; ++o) best[o] = 0ull;

    for (int p = tid; p < NPRI; p += blockDim.x) {
        float pcx = pboxes[p*4+0], pcy = pboxes[p*4+1];
        float pw  = pboxes[p*4+2], ph  = pboxes[p*4+3];
        float bx0 = pcx - pw*0.5f, by0 = pcy - ph*0.5f;
        float bx1 = pcx + pw*0.5f, by1 = pcy + ph*0.5f;
        float areaB = (bx1 - bx0) * (by1 - by0);
        float mx = -1.0f; int mo = 0;
        #pragma unroll
        for (int o = 0; o < NOBJ; ++o) {
            float lx = fmaxf(s_bc[o][0], bx0), ly = fmaxf(s_bc[o][1], by0);
            float rx = fminf(s_bc[o][2], bx1), ry = fminf(s_bc[o][3], by1);
            float iw = fmaxf(rx - lx, 0.0f),  ih = fmaxf(ry - ly, 0.0f);
            float inter = iw * ih;
            float areaA = (s_bc[o][2]-s_bc[o][0]) * (s_bc[o][3]-s_bc[o][1]);
            float iou = inter / (areaA + areaB - inter);
            if (iou > mx) { mx = iou; mo = o; }   // strict > keeps lowest index (argmax tie rule)
            unsigned long long pk =
                ((unsigned long long)__float_as_uint(iou) << 32) |
                (unsigned long long)(0xFFFFFFFFu - (unsigned)p);   // lower prior wins ties
            if (pk > best[o]) best[o] = pk;
        }
        ov_g [b * NPRI + p] = mx;
        idx_g[b * NPRI + p] = mo;
    }
    #pragma unroll
    for (int o = 0; o < NOBJ; ++o) atomicMax(&s_best[o], best[o]);

    __threadfence();
    __syncthreads();

    if (tid == 0) {   // sequential: matches scatter "last wins" on duplicate best priors
        for (int o = 0; o < NOBJ; ++o) {
            unsigned bp = 0xFFFFFFFFu - (unsigned)(s_best[o] & 0xFFFFFFFFull);
            idx_g[b * NPRI + bp] = o;
            ov_g [b * NPRI + bp] = 1.0f;
        }
    }
    __threadfence();
    __syncthreads();

    float absum = 0.0f; int cnt = 0;
    for (int p = tid; p < NPRI; p += blockDim.x) {
        float ovp = ov_g [b * NPRI + p];
        int   oi  = idx_g[b * NPRI + p];
        int   lab = (ovp < 0.5f) ? 0 : s_tc[oi];
        labels_g[b * NPRI + p] = lab;
        if (lab != 0) {
            float pcx = pboxes[p*4+0], pcy = pboxes[p*4+1];
            float pw  = pboxes[p*4+2], ph  = pboxes[p*4+3];
            float gx = (s_cc[oi][0] - pcx) / (pw * 0.1f);
            float gy = (s_cc[oi][1] - pcy) / (ph * 0.1f);
            float gw = logf(s_cc[oi][2] / pw) * 5.0f;
            float gh = logf(s_cc[oi][3] / ph) * 5.0f;
            const float* pb = pred_boxes + ((size_t)(b * NPRI + p)) * 4;
            absum += fabsf(pb[0]-gx) + fabsf(pb[1]-gy) + fabsf(pb[2]-gw) + fabsf(pb[3]-gh);
            cnt++;
        }
    }
    s_rf[tid] = absum; s_ri[tid] = cnt;
    __syncthreads();
    for (int s = 128; s > 0; s >>= 1) {
        if (tid < s) { s_rf[tid] += s_rf[tid + s]; s_ri[tid] += s_ri[tid + s]; }
        __syncthreads();
    }
    if (tid == 0) {
        atomicAdd(&fsc[0], s_rf[0]);
        atomicAdd(usc, (unsigned)s_ri[0]);
    }
}

// ---------------------------------------------------------------- per-row CE (WMMA exp-sum)
__global__ void __launch_bounds__(256)
k_cls(const float* __restrict__ pred_scores,
      const int*   __restrict__ labels_g,
      float* __restrict__ cls_all,
      float* __restrict__ cls_pos_g)
{
    __shared__ float smem[8][16 * 84];   // 8 waves, 16 rows x 84 (81 classes + pad)
    const int lane = threadIdx.x & 31;
    const int wv   = threadIdx.x >> 5;
    const int g    = blockIdx.x * 8 + wv;        // row group; rows g*16 .. g*16+15
    const int base = g * 16 * NCLS;

    // coalesced stage of 16 rows into LDS (same-wave producer/consumer, DS in-order)
    for (int i = lane; i < 16 * NCLS; i += 32) {
        int r = i / NCLS;
        int c = i - r * NCLS;
        smem[wv][r * 84 + c] = pred_scores[base + i];
    }

    const int mrow = lane & 15;
    const int half = lane >> 4;
    const float* rowp = &smem[wv][mrow * 84];

    // per-row max: lane pair (mrow, mrow+16) each scans half of each 4-chunk
    float m = -3.4e38f;
    #pragma unroll
    for (int j = 0; j < 21; ++j) {
        int c0 = 4 * j + 2 * half;
        if (c0     < NCLS) m = fmaxf(m, rowp[c0]);
        if (c0 + 1 < NCLS) m = fmaxf(m, rowp[c0 + 1]);
    }
    m = fmaxf(m, __shfl_xor(m, 16, 32));

    // sum_c exp(x-m) via f32 WMMA: D = A(16x4 of exp values) x B(4x16 ones), exact fp32
    float S;
#if __has_builtin(__builtin_amdgcn_wmma_f32_16x16x4_f32)
    v8f acc = {0.f,0.f,0.f,0.f,0.f,0.f,0.f,0.f};
    v2f ones; ones.x = 1.0f; ones.y = 1.0f;
    #pragma unroll
    for (int j = 0; j < 21; ++j) {
        int c0 = 4 * j + 2 * half;
        v2f a;
        a.x = (c0     < NCLS) ? expf(rowp[c0]     - m) : 0.0f;
        a.y = (c0 + 1 < NCLS) ? expf(rowp[c0 + 1] - m) : 0.0f;
        acc = __builtin_amdgcn_wmma_f32_16x16x4_f32(
            false, a, false, ones, (short)0, acc, false, false);
    }
    // D layout: vgpr v, lanes 0-15 -> (M=v, N=lane); lanes 16-31 -> (M=v+8)
    int e = lane & 7;
    float X = acc[0];
    X = (e == 1) ? acc[1] : X;
    X = (e == 2) ? acc[2] : X;
    X = (e == 3) ? acc[3] : X;
    X = (e == 4) ? acc[4] : X;
    X = (e == 5) ? acc[5] : X;
    X = (e == 6) ? acc[6] : X;
    X = (e == 7) ? acc[7] : X;
    int src = (mrow < 8) ? mrow : (mrow + 8);   // rows 8..15 live in lanes 16..23
    S = __shfl(X, src, 32);
#else
    float ps = 0.0f;
    #pragma unroll
    for (int j = 0; j < 21; ++j) {
        int c0 = 4 * j + 2 * half;
        if (c0     < NCLS) ps += expf(rowp[c0]     - m);
        if (c0 + 1 < NCLS) ps += expf(rowp[c0 + 1] - m);
    }
    S = ps + __shfl_xor(ps, 16, 32);
#endif

    float contrib = 0.0f;
    if (lane < 16) {
        int rid = g * 16 + mrow;
        int lab = labels_g[rid];
        float st  = rowp[lab];
        float ce  = (m + logf(S)) - st;    // -log_softmax[true class]
        cls_all[rid] = ce;
        if (lab != 0) contrib = ce;
    }
    contrib += __shfl_xor(contrib, 16, 32);
    contrib += __shfl_xor(contrib,  8, 32);
    contrib += __shfl_xor(contrib,  4, 32);
    contrib += __shfl_xor(contrib,  2, 32);
    contrib += __shfl_xor(contrib,  1, 32);
    if (lane == 0) atomicAdd(cls_pos_g, contrib);
}

// ---------------------------------------------------------------- hard-negative top-K sum
__global__ void __launch_bounds__(1024)
k_hard(const float* __restrict__ cls_all,
       const unsigned* __restrict__ npos_g,
       float* __restrict__ cls_hard_g)
{
    __shared__ float s[SORT_N];          // 64 KB (gfx1250: 320 KB/WGP)
    const int b   = blockIdx.x;
    const int tid = threadIdx.x;
    const int nt  = blockDim.x;

    unsigned npos = *npos_g;
    unsigned K = 3u * npos;
    if (K > NPRI) K = NPRI;              // mask i < 3*n_pos, clamped to P

    for (int i = tid; i < SORT_N; i += nt)
        s[i] = (i < NPRI) ? cls_all[b * NPRI + i] : -3.4e38f;
    __syncthreads();

    // bitonic sort, descending
    for (unsigned k = 2; k <= SORT_N; k <<= 1) {
        for (unsigned j = k >> 1; j > 0; j >>= 1) {
            for (unsigned t = tid; t < SORT_N / 2; t += nt) {
                unsigned i   = 2u * t - (t & (j - 1u));
                unsigned ixj = i + j;
                float a = s[i], c = s[ixj];
                bool desc = ((i & k) == 0u);
                if ((a < c) == desc) { s[i] = c; s[ixj] = a; }
            }
            __syncthreads();
        }
    }

    float sum = 0.0f;
    for (unsigned i = tid; i < K; i += nt) sum += s[i];
    sum += __shfl_xor(sum, 16, 32);
    sum += __shfl_xor(sum,  8, 32);
    sum += __shfl_xor(sum,  4, 32);
    sum += __shfl_xor(sum,  2, 32);
    sum += __shfl_xor(sum,  1, 32);
    if ((tid & 31) == 0) atomicAdd(cls_hard_g, sum);
}

// ---------------------------------------------------------------- finalize
__global__ void k_final(const float* fsc, const unsigned* usc, float* out) {
    if (threadIdx.x == 0 && blockIdx.x == 0) {
        float npos = (float)usc[0];
        float loc_loss = fsc[0] / (npos * 4.0f);
        float cls_loss = (fsc[1] + fsc[2]) / npos;
        out[0] = loc_loss + cls_loss;
    }
}

// ---------------------------------------------------------------- launch
extern "C" void kernel_launch(void* const* d_in, const int* in_sizes, int n_in,
                              void* d_out, int out_size, void* d_ws, size_t ws_size,
                              hipStream_t stream)
{
    const float* pred_boxes   = (const float*)d_in[0];
    const float* pred_scores  = (const float*)d_in[1];
    const float* true_boxes   = (const float*)d_in[2];
    const int*   true_classes = (const int*)  d_in[3];
    const float* pboxes       = (const float*)d_in[4];

    char* ws = (char*)d_ws;
    const size_t N4 = (size_t)NROWS * 4;
    int*      labels  = (int*)   (ws);
    float*    cls_all = (float*) (ws + 1 * N4);
    float*    ov      = (float*) (ws + 2 * N4);
    int*      idx     = (int*)   (ws + 3 * N4);
    float*    fsc     = (float*) (ws + 4 * N4);        // [0]=abs,[1]=cls_pos,[2]=cls_hard
    unsigned* usc     = (unsigned*)(ws + 4 * N4 + 12); // n_pos

    k_init <<<1, 64, 0, stream>>>(fsc, usc);
    k_match<<<NB, 256, 0, stream>>>(pred_boxes, true_boxes, true_classes, pboxes,
                                    ov, idx, labels, fsc, usc);
    k_cls  <<<CLS_BLOCKS, 256, 0, stream>>>(pred_scores, labels, cls_all, &fsc[1]);
    k_hard <<<NB, 1024, 0, stream>>>(cls_all, usc, &fsc[2]);
    k_final<<<1, 64, 0, stream>>>(fsc, usc, (float*)d_out);
}